// GreedySampler_76484777607854
// MI455X (gfx1250) — compile-verified
//
#include <hip/hip_runtime.h>
#include <hip/hip_bf16.h>

// GreedySampler on MI455X (gfx1250):
//   M=200 (pad 256) x N=50257 x K=4096 f32 GEMM fused with row argmax,
//   using V_WMMA_F32_16X16X4_F32 (full-precision CDNA5 WMMA) so the argmax
//   is bit-faithful to the f32 reference. Weights stream once from HBM
//   (823 MB ~ 35us floor at 23.3 TB/s) with non-temporal hints; the 3.3 MB
//   gathered A matrix stays hot in the 192 MB L2, so no LDS staging needed.

#define D_MODEL 4096
#define VOCAB   50257
#define M_PAD   256      // 16 tiles of 16 rows (200 real rows, rest zero)

typedef float v2f __attribute__((ext_vector_type(2)));
typedef float v8f __attribute__((ext_vector_type(8)));

// ---------------------------------------------------------------------------
// Kernel 1: gather the 200 job rows into ws (padded to 256 rows, zero pad),
// and initialize the per-row argmax slots.
// ---------------------------------------------------------------------------
__global__ __launch_bounds__(256)
void gs_gather_init(const float* __restrict__ hs,
                    const int*   __restrict__ fill,      // [n_fill]
                    const int*   __restrict__ ngen_ptr,  // scalar on device
                    int n_fill,
                    float* __restrict__ Aws,             // [M_PAD][D_MODEL]
                    unsigned long long* __restrict__ slots) {
  const int row = blockIdx.x;           // 0..255 padded row
  const int t   = threadIdx.x;          // 0..255

  // Source row index (cumsum computed inline; tiny, cached reads).
  int src = -1;
  if (row < n_fill) {
    int c = 0;
    for (int i = 0; i <= row; ++i) c += fill[i];
    src = c - 1;                                       // last token of prefill
  } else {
    int total = 0;
    for (int i = 0; i < n_fill; ++i) total += fill[i];
    int g = row - n_fill;
    if (g < *ngen_ptr) src = total + g;                // decode token
  }

  float4* d = (float4*)(Aws + (size_t)row * D_MODEL);
  if (src >= 0) {
    const float4* s = (const float4*)(hs + (size_t)src * D_MODEL);
#pragma unroll
    for (int i = 0; i < 4; ++i) d[t * 4 + i] = s[t * 4 + i];
  } else {
    float4 z = make_float4(0.f, 0.f, 0.f, 0.f);
#pragma unroll
    for (int i = 0; i < 4; ++i) d[t * 4 + i] = z;
  }
  if (t == 0) slots[row] = 0ull;   // minimal packed (value,col) key
}

// Order-preserving f32 -> u32 map (monotone for all finite floats).
__device__ __forceinline__ unsigned int ordered_bits(float v) {
  unsigned int u = __float_as_uint(v);
  return (u & 0x80000000u) ? ~u : (u | 0x80000000u);
}

// ---------------------------------------------------------------------------
// Kernel 2: fused GEMM + argmax.
//   block = 128 threads = 4 waves. Wave w owns rows [64w, 64w+64) (4 m-tiles)
//   for this block's 64 vocab columns (4 n-tiles). 16 accumulator tiles/wave.
//   V_WMMA_F32_16X16X4_F32 fragment layout (ISA 7.12.2):
//     A 16x4: lane&15 = M row; VGPR pair holds K = 2*(lane>>4) .. +1
//     B 4x16: lane&15 = N col; VGPR pair holds K = 2*(lane>>4) .. +1
//     C 16x16: VGPR v, lanes 0-15 -> M=v, lanes 16-31 -> M=v+8; N = lane&15
//   B fragments use non-temporal loads: the 823 MB weight stream is
//   read-once and must not evict the L2-resident A matrix.
// ---------------------------------------------------------------------------
__global__ __launch_bounds__(128)
void gs_gemm_argmax(const float* __restrict__ Aws,   // [M_PAD][D_MODEL]
                    const float* __restrict__ W,     // [VOCAB][D_MODEL]
                    unsigned long long* __restrict__ slots) {
  const int lane = threadIdx.x & 31;
  const int wave = threadIdx.x >> 5;        // 0..3
  const int rl   = lane & 15;               // row/col selector within tile
  const int h    = lane >> 4;               // K half: holds K = 2h, 2h+1
  const int m_base = wave * 64;
  const int n_base = blockIdx.x * 64;

  // Column per n-tile, clamped so the ragged last block stays in bounds.
  // Clamped duplicates pack an identical (value, col) key -> argmax unchanged.
  int col[4];
#pragma unroll
  for (int nt = 0; nt < 4; ++nt) {
    int c = n_base + nt * 16 + rl;
    col[nt] = c < (VOCAB - 1) ? c : (VOCAB - 1);
  }

  const float* aptr = Aws + (size_t)(m_base + rl) * D_MODEL + 2 * h;
  const float* bptr[4];
#pragma unroll
  for (int nt = 0; nt < 4; ++nt)
    bptr[nt] = W + (size_t)col[nt] * D_MODEL + 2 * h;

  v8f acc[4][4];
#pragma unroll
  for (int mt = 0; mt < 4; ++mt)
#pragma unroll
    for (int nt = 0; nt < 4; ++nt)
      acc[mt][nt] = (v8f){0.f, 0.f, 0.f, 0.f, 0.f, 0.f, 0.f, 0.f};

  for (int k = 0; k < D_MODEL; k += 4) {
    v2f a[4], b[4];
#pragma unroll
    for (int mt = 0; mt < 4; ++mt)
      a[mt] = *(const v2f*)(aptr + (size_t)mt * 16 * D_MODEL + k);
#pragma unroll
    for (int nt = 0; nt < 4; ++nt)
      b[nt] = __builtin_nontemporal_load((const v2f*)(bptr[nt] + k));
#pragma unroll
    for (int mt = 0; mt < 4; ++mt)
#pragma unroll
      for (int nt = 0; nt < 4; ++nt)
        acc[mt][nt] = __builtin_amdgcn_wmma_f32_16x16x4_f32(
            false, a[mt], false, b[nt], (short)0, acc[mt][nt], false, false);
  }

  // Per-row argmax over this wave's 64 columns, then atomic-max into slots.
  // Row m = m_base + mt*16 + v + 8h lives in acc[mt][*][v] at column col[nt].
#pragma unroll
  for (int mt = 0; mt < 4; ++mt) {
#pragma unroll
    for (int v = 0; v < 8; ++v) {
      float best = acc[mt][0][v];
      int   bcol = col[0];
#pragma unroll
      for (int nt = 1; nt < 4; ++nt) {
        float f = acc[mt][nt][v];
        if (f > best) { best = f; bcol = col[nt]; }   // '>' keeps smaller col on tie
      }
      // key: high 32 = ordered value bits, low 32 = ~col (tie -> smaller col).
      unsigned long long key =
          ((unsigned long long)ordered_bits(best) << 32) |
          (unsigned long long)(unsigned int)(~(unsigned int)bcol);
      // Reduce over the 16 lanes of this half (xor masks 1..8 stay in-half).
#pragma unroll
      for (int m = 8; m >= 1; m >>= 1) {
        unsigned long long o = __shfl_xor(key, m, 32);
        if (o > key) key = o;
      }
      if (rl == 0) {
        int row = m_base + mt * 16 + v + 8 * h;
        atomicMax(&slots[row], key);
      }
    }
  }
}

// ---------------------------------------------------------------------------
// Kernel 3: unpack winning columns.
// ---------------------------------------------------------------------------
__global__ __launch_bounds__(256)
void gs_finalize(const unsigned long long* __restrict__ slots,
                 const int* __restrict__ ngen_ptr, int n_fill,
                 float* __restrict__ out, int out_size) {
  int m = blockIdx.x * blockDim.x + threadIdx.x;
  int n_jobs = n_fill + *ngen_ptr;
  if (n_jobs > out_size) n_jobs = out_size;
  if (m < n_jobs) {
    unsigned int inv = (unsigned int)(slots[m] & 0xffffffffull);
    out[m] = (float)(int)(~inv);
  }
}

extern "C" void kernel_launch(void* const* d_in, const int* in_sizes, int n_in,
                              void* d_out, int out_size, void* d_ws, size_t ws_size,
                              hipStream_t stream) {
  const float* hs   = (const float*)d_in[0];   // [8192, 4096] f32
  const float* W    = (const float*)d_in[1];   // [50257, 4096] f32
  const int*   fill = (const int*)d_in[2];     // [8] i32
  const int*   ngen = (const int*)d_in[3];     // scalar i32 (device)
  const int n_fill  = in_sizes[2];

  float* out = (float*)d_out;

  // Workspace: [0, 4MiB) packed A, then 256 u64 argmax slots.
  float* Aws = (float*)d_ws;
  unsigned long long* slots =
      (unsigned long long*)((char*)d_ws + (size_t)M_PAD * D_MODEL * sizeof(float));

  gs_gather_init<<<M_PAD, 256, 0, stream>>>(hs, fill, ngen, n_fill, Aws, slots);

  const int n_blocks = (VOCAB + 63) / 64;      // 786
  gs_gemm_argmax<<<n_blocks, 128, 0, stream>>>(Aws, W, slots);

  gs_finalize<<<1, 256, 0, stream>>>(slots, ngen, n_fill, out, out_size);
}